// CRFLayer_41695542509984
// MI455X (gfx1250) — compile-verified
//
#include <hip/hip_runtime.h>
#include <hip/hip_bf16.h>
#include <math.h>

// Problem dims (fixed by the reference)
#define B_DIM 64
#define T_DIM 512
#define D_DIM 768
#define L_DIM 50
#define NPAD  64          // L padded to 64 for 4 WMMA N-tiles
#define BT    (B_DIM * T_DIM)

typedef __attribute__((ext_vector_type(2))) float v2f;
typedef __attribute__((ext_vector_type(8))) float v8f;

// ---------------------------------------------------------------------------
// Kernel 0: pad W [768,50] -> Wp [768,64] (zero-filled cols 50..63)
// ---------------------------------------------------------------------------
__global__ void wpad_kernel(const float* __restrict__ W, float* __restrict__ Wp) {
    int idx = blockIdx.x * 256 + threadIdx.x;       // 768*64 = 49152 elems
    if (idx < D_DIM * NPAD) {
        int k = idx >> 6;
        int n = idx & 63;
        Wp[idx] = (n < L_DIM) ? W[k * L_DIM + n] : 0.0f;
    }
}

// ---------------------------------------------------------------------------
// Kernel 1: logits = A @ Wp + bias via V_WMMA_F32_16X16X4_F32.
// One wave per 16-row M tile; 4 N-tiles (64 cols) accumulated per wave so
// each A fragment is reused 4x. 192 K-steps * 4 wmma = 768 wmma / wave.
// Memory-bound on A (~100 MB streamed once); W stays hot in the 192 MB L2.
// ---------------------------------------------------------------------------
__global__ void __launch_bounds__(32)
gemm_wmma_kernel(const float* __restrict__ A,      // [BT, 768]
                 const float* __restrict__ Wp,     // [768, 64]
                 const float* __restrict__ bias,   // [50]
                 float* __restrict__ logits)       // [BT, 50]
{
    const int mtile = blockIdx.x;                  // 0..2047
    const int lane  = threadIdx.x;                 // 0..31
    const int half  = lane >> 4;                   // 0 or 1
    const int l     = lane & 15;

    const float* arow = A + (size_t)(mtile * 16 + l) * D_DIM;

    v8f acc[4];
#pragma unroll
    for (int nt = 0; nt < 4; ++nt)
        acc[nt] = (v8f){0.f, 0.f, 0.f, 0.f, 0.f, 0.f, 0.f, 0.f};

    for (int k = 0; k < D_DIM; k += 4) {
        // A fragment: contiguous pair A[row][k + 2*half .. +1]
        v2f a = *(const v2f*)(arow + k + 2 * half);
        const float* bp = Wp + (size_t)(k + 2 * half) * NPAD + l;
#pragma unroll
        for (int nt = 0; nt < 4; ++nt) {
            v2f bf;
            bf.x = bp[nt * 16];           // K = k+2*half,   N = nt*16 + l
            bf.y = bp[nt * 16 + NPAD];    // K = k+2*half+1
            acc[nt] = __builtin_amdgcn_wmma_f32_16x16x4_f32(
                /*neg_a=*/false, a, /*neg_b=*/false, bf,
                /*c_mod=*/(short)0, acc[nt],
                /*reuse_a=*/false, /*reuse_b=*/false);
        }
    }

    // Epilogue: bias add + masked store (N < 50)
#pragma unroll
    for (int nt = 0; nt < 4; ++nt) {
        int col = nt * 16 + l;
        if (col < L_DIM) {
            float bv = bias[col];
#pragma unroll
            for (int r = 0; r < 8; ++r) {
                int row = mtile * 16 + half * 8 + r;
                logits[(size_t)row * L_DIM + col] = acc[nt][r] + bv;
            }
        }
    }
}

// ---------------------------------------------------------------------------
// Kernel 2: CRF log-likelihood per batch element (one block = 2 waves).
//
// Scaled forward algorithm: with M = max_i alpha_i and E = exp(trans)
// (precomputed once in LDS),
//   nxt_j = M + log( sum_i exp(alpha_i - M) * E_ij ) + em[t,j]
// so the per-step inner loop is 50 FMAs; one exp + one log per thread/step.
// alpha_j lives in a register (thread j only ever reads its own alpha).
// M maintained via wave32 shfl_xor max-reduce + 2-wave LDS combine.
//
// Numerator parallelized over t (mask is a contiguous valid prefix, so
// prev_tag(t) == tags[t-1] inside the valid region); the last valid tag is
// found at the mask falling edge. Strided accumulation + block reduction.
// ---------------------------------------------------------------------------
__global__ void __launch_bounds__(64)
crf_kernel(const float* __restrict__ logits,       // [B, T, L]
           const int* __restrict__ labels,         // [B, T]
           const unsigned char* __restrict__ mask, // [B, T] (bool bytes)
           const float* __restrict__ start_t,      // [L]
           const float* __restrict__ end_t,        // [L]
           const float* __restrict__ trans,        // [L, L]
           float* __restrict__ llh_out)            // [B]
{
    __shared__ float trans_s[L_DIM * L_DIM];  // raw transitions (numerator)
    __shared__ float E_s[L_DIM * L_DIM];      // exp(transitions) (denominator)
    __shared__ float p_s[L_DIM];              // exp(alpha - M)
    __shared__ float wred[2];                 // per-wave max
    __shared__ float wsum[2];                 // per-wave sum
    __shared__ int   ltag_s[2];               // per-wave last-tag
    __shared__ float num_s;

    const int b    = blockIdx.x;
    const int tid  = threadIdx.x;
    const int wid  = tid >> 5;
    const int lane = tid & 31;

    const float*         em = logits + (size_t)b * T_DIM * L_DIM;
    const int*           lb = labels + (size_t)b * T_DIM;
    const unsigned char* mk = mask   + (size_t)b * T_DIM;

    // Stage transitions + exp(transitions) into LDS
    for (int i = tid; i < L_DIM * L_DIM; i += 64) {
        float v = trans[i];
        trans_s[i] = v;
        E_s[i]     = __expf(v);
    }
    __syncthreads();

    // ---- numerator: parallel over t ----
    float part = 0.0f;
    int last_tag = -1;
    if (tid == 0) {
        int t0 = lb[0];
        part += start_t[t0] + em[t0];
        if (!mk[1]) last_tag = t0;                 // length-1 sequence edge case
    }
    for (int t = 1 + tid; t < T_DIM; t += 64) {
        if (mk[t]) {
            int tg = lb[t];
            part += trans_s[lb[t - 1] * L_DIM + tg] + em[(size_t)t * L_DIM + tg];
            if (t == T_DIM - 1 || !mk[t + 1]) last_tag = tg;  // mask falling edge
        }
    }
#pragma unroll
    for (int off = 16; off > 0; off >>= 1) {
        part += __shfl_xor(part, off);
        last_tag = max(last_tag, __shfl_xor(last_tag, off));
    }
    if (lane == 0) { wsum[wid] = part; ltag_s[wid] = last_tag; }
    __syncthreads();
    if (tid == 0)
        num_s = wsum[0] + wsum[1] + end_t[max(ltag_s[0], ltag_s[1])];
    __syncthreads();

    // ---- denominator: scaled forward algorithm ----
    float alpha = 0.0f;
    if (tid < L_DIM) alpha = start_t[tid] + em[tid];

    // initial running max M
    float m0 = (tid < L_DIM) ? alpha : -INFINITY;
#pragma unroll
    for (int off = 16; off > 0; off >>= 1)
        m0 = fmaxf(m0, __shfl_xor(m0, off));
    if (lane == 0) wred[wid] = m0;
    __syncthreads();
    float M = fmaxf(wred[0], wred[1]);

    for (int t = 1; t < T_DIM; ++t) {
        if (mk[t]) {                               // uniform across the block
            if (tid < L_DIM) p_s[tid] = __expf(alpha - M);
            __syncthreads();
            if (tid < L_DIM) {
                float s = 0.0f;
#pragma unroll
                for (int i = 0; i < L_DIM; ++i)
                    s = fmaf(p_s[i], E_s[i * L_DIM + tid], s);
                alpha = M + __logf(s) + em[(size_t)t * L_DIM + tid];
            }
            // refresh M = max_j alpha_j
            float m = (tid < L_DIM) ? alpha : -INFINITY;
#pragma unroll
            for (int off = 16; off > 0; off >>= 1)
                m = fmaxf(m, __shfl_xor(m, off));
            if (lane == 0) wred[wid] = m;
            __syncthreads();                       // also guards p_s reuse
            M = fmaxf(wred[0], wred[1]);
        }
    }

    // ---- den = logsumexp(alpha + end), llh = num - den ----
    float v  = (tid < L_DIM) ? alpha + end_t[tid] : -INFINITY;
    float mm = v;
#pragma unroll
    for (int off = 16; off > 0; off >>= 1)
        mm = fmaxf(mm, __shfl_xor(mm, off));
    if (lane == 0) wred[wid] = mm;
    __syncthreads();
    mm = fmaxf(wred[0], wred[1]);

    float e = (tid < L_DIM) ? __expf(v - mm) : 0.0f;
#pragma unroll
    for (int off = 16; off > 0; off >>= 1)
        e += __shfl_xor(e, off);
    if (lane == 0) wsum[wid] = e;
    __syncthreads();
    if (tid == 0) {
        float den = mm + __logf(wsum[0] + wsum[1]);
        llh_out[b] = num_s - den;
    }
}

// ---------------------------------------------------------------------------
// Kernel 3: loss = -mean(llh) -> d_out[0]
// ---------------------------------------------------------------------------
__global__ void reduce_kernel(const float* __restrict__ llh, float* __restrict__ out) {
    if (blockIdx.x == 0 && threadIdx.x == 0) {
        float s = 0.0f;
        for (int i = 0; i < B_DIM; ++i) s += llh[i];
        out[0] = -(s / (float)B_DIM);
    }
}

// ---------------------------------------------------------------------------
// Host launcher
// Inputs (setup_inputs order):
//   0 input_emb f32 [B,T,D]   1 labels i32 [B,T]   2 label_mask bool [B,T]
//   3 W f32 [D,L]  4 b f32 [L]  5 start_t f32 [L]  6 end_t f32 [L]  7 trans f32 [L,L]
// Output: d_out[0] = loss scalar, d_out[1..] = logits [B,T,L] flat.
// ---------------------------------------------------------------------------
extern "C" void kernel_launch(void* const* d_in, const int* in_sizes, int n_in,
                              void* d_out, int out_size, void* d_ws, size_t ws_size,
                              hipStream_t stream) {
    (void)in_sizes; (void)n_in; (void)out_size; (void)ws_size;

    const float*         emb     = (const float*)d_in[0];
    const int*           labels  = (const int*)d_in[1];
    const unsigned char* lmask   = (const unsigned char*)d_in[2]; // jnp bool = 1 byte
    const float*         W       = (const float*)d_in[3];
    const float*         bias    = (const float*)d_in[4];
    const float*         start_t = (const float*)d_in[5];
    const float*         end_t   = (const float*)d_in[6];
    const float*         trans   = (const float*)d_in[7];

    float* out    = (float*)d_out;
    float* logits = out + 1;                   // [B*T, L]

    float* Wp  = (float*)d_ws;                 // 768*64 floats
    float* llh = Wp + D_DIM * NPAD;            // 64 floats

    // 0) pad W to [768,64]
    wpad_kernel<<<(D_DIM * NPAD + 255) / 256, 256, 0, stream>>>(W, Wp);

    // 1) WMMA GEMM: 2048 M-tiles, one wave each
    gemm_wmma_kernel<<<BT / 16, 32, 0, stream>>>(emb, Wp, bias, logits);

    // 2) CRF scan: one block per batch element
    crf_kernel<<<B_DIM, 64, 0, stream>>>(logits, labels, lmask, start_t,
                                         end_t, trans, llh);

    // 3) final scalar
    reduce_kernel<<<1, 32, 0, stream>>>(llh, out);
}